// Heston_51848845197666
// MI455X (gfx1250) — compile-verified
//
#include <hip/hip_runtime.h>

#define AS3 __attribute__((address_space(3)))

namespace {

constexpr int kBatch  = 65536;
constexpr int kSteps  = 512;
constexpr int kTK     = 16;                 // steps per LDS tile
constexpr int kTiles  = kSteps / kTK;       // 32
constexpr int kZpStride = kTK + 4;          // 20 floats/path row (pad -> 2-way conflict max)
constexpr int kZvStride = kTK + 4;          // 20 floats/path row (comp0 only, compacted)
constexpr int kZpBuf  = 32 * kZpStride;     // 640 floats per buffer per wave
constexpr int kZvBuf  = 32 * kZvStride;     // 640 floats per buffer per wave
constexpr int kBufFloats = kZpBuf + kZvBuf; // 1280 floats = 5120 B
constexpr int kWaves  = 2;                  // waves per block
constexpr int kBlock  = 32 * kWaves;        // 64 threads -> 20,480 B LDS/block

// Heston constants (double-accurate literals, rounded to fp32 at compile time)
constexpr float kDT       = 0.001953125f;             // 1/512
constexpr float kV0       = 0.055225f;                // 0.235^2
constexpr float kSigvSqdt = 0.08838834764831843f;     // sigma_v * sqrt(dt)
constexpr float kC1       = -0.0014088819735243784f;  // -0.5*dt*log2(e)
constexpr float kC2       = -0.0446311006f;           // rho*sqrt(dt)*log2(e)
constexpr float kC3       =  0.0455328300f;           // rho_perp*sqrt(dt)*log2(e)

__device__ __forceinline__ unsigned ldsOffset(const void* p) {
  return (unsigned)(unsigned long long)(AS3 const void*)p;
}

// Async DMA: 32 lanes x 16B contiguous global -> per-lane LDS addresses.
__device__ __forceinline__ void asyncCopyB128(unsigned ldsByteAddr, const float* gptr) {
  asm volatile("global_load_async_to_lds_b128 %0, %1, off"
               :: "v"(ldsByteAddr), "v"(gptr)
               : "memory");
}

// Async DMA gather: 32 lanes x 4B (per-lane global addresses) -> LDS.
// Used to strip the dead Z_vol[...,1] component while compacting into LDS.
__device__ __forceinline__ void asyncCopyB32(unsigned ldsByteAddr, const float* gptr) {
  asm volatile("global_load_async_to_lds_b32 %0, %1, off"
               :: "v"(ldsByteAddr), "v"(gptr)
               : "memory");
}

// Stage one [32 paths x kTK steps] tile of Zp and Zv(comp0) into LDS.
// Zp row = 64B contiguous -> 4 lanes/row, 8 rows/instr, 4 x b128.
// Zv row = 16 comp0 elems (global stride 8B) -> 16 lanes/row, 2 rows/instr, 16 x b32.
// 20 async copies total per tile.
__device__ __forceinline__ void fillTile(const float* __restrict__ Zv,
                                         const float* __restrict__ Zp,
                                         unsigned zpLds, unsigned zvLds,
                                         int pbase, int k0, int lane) {
  {
    const int row = lane >> 2, chunk = lane & 3;
    const float* g = Zp + (size_t)(pbase + row) * kSteps + k0 + chunk * 4;
    const unsigned l = zpLds + (unsigned)(row * kZpStride + chunk * 4) * 4u;
#pragma unroll
    for (int j = 0; j < 4; ++j)
      asyncCopyB128(l + (unsigned)(j * 8 * kZpStride * 4), g + (size_t)j * 8 * kSteps);
  }
  {
    const int row = lane >> 4, s = lane & 15;
    const float* g = Zv + ((size_t)(pbase + row) * kSteps + k0 + s) * 2;  // comp0 only
    const unsigned l = zvLds + (unsigned)(row * kZvStride + s) * 4u;
#pragma unroll
    for (int j = 0; j < 16; ++j)
      asyncCopyB32(l + (unsigned)(j * 2 * kZvStride * 4), g + (size_t)j * 2 * kSteps * 2);
  }
}

} // namespace

__global__ __launch_bounds__(kBlock) void heston_paths_kernel(
    const float* __restrict__ Zv, const float* __restrict__ Zp,
    float* __restrict__ out) {
  __shared__ float smem[kWaves * 2 * kBufFloats];  // 20,480 bytes

  const int lane  = threadIdx.x & 31;
  const int wave  = threadIdx.x >> 5;
  const int pbase = ((int)blockIdx.x * kWaves + wave) * 32;
  const int path  = pbase + lane;

  float* __restrict__ Sout = out + (size_t)path * (kSteps + 1);
  float* __restrict__ Vout = out + (size_t)kBatch * (kSteps + 1) + (size_t)path * (kSteps + 1);
  Sout[0] = 100.0f;
  Vout[0] = kV0;

  float* waveSmem = &smem[wave * 2 * kBufFloats];
  const unsigned waveLds = ldsOffset(waveSmem);

  // Prime the pipeline: tile 0 -> buffer 0.
  fillTile(Zv, Zp, waveLds, waveLds + kZpBuf * 4, pbase, 0, lane);

  float V = kV0;
  float logS2 = 0.0f;  // log2(S / S0)

  for (int t = 0; t < kTiles; ++t) {
    if (t + 1 < kTiles) {
      const unsigned base = waveLds + (unsigned)(((t + 1) & 1) * kBufFloats * 4);
      fillTile(Zv, Zp, base, base + kZpBuf * 4, pbase, (t + 1) * kTK, lane);
      // 40 outstanding; async loads complete in order, so draining to 20
      // guarantees tile t has fully landed in LDS.
      asm volatile("s_wait_asynccnt 20" ::: "memory");
    } else {
      asm volatile("s_wait_asynccnt 0" ::: "memory");
    }

    const float* zpT = waveSmem + (t & 1) * kBufFloats + lane * kZpStride;
    const float* zvT = waveSmem + (t & 1) * kBufFloats + kZpBuf + lane * kZvStride;

    float* __restrict__ So = Sout + t * kTK + 1;
    float* __restrict__ Vo = Vout + t * kTK + 1;

#pragma unroll
    for (int k = 0; k < kTK; ++k) {
      const float zp = zpT[k];
      const float zv = zvT[k];

      const float vp = fmaxf(V, 0.0f);                  // full truncation
      const float sq = __builtin_amdgcn_sqrtf(vp);      // v_sqrt_f32
      float Vn = fmaf(kSigvSqdt * zv, sq, V);           // + sigma*sqrt(vp*dt)*zv
      Vn = fmaf(0.04f - vp, kDT, Vn);                   // + kappa*(theta-vp)*dt
      Vn = fmaxf(Vn, 0.0f);

      const float dB2 = fmaf(kC2, zv, kC3 * zp);        // log2e-scaled Brownian inc
      logS2 = fmaf(kC1, vp, fmaf(sq, dB2, logS2));      // cumulated log2 price
      So[k] = 100.0f * __builtin_amdgcn_exp2f(logS2);   // v_exp_f32
      Vo[k] = Vn;
      V = Vn;
    }
  }
}

extern "C" void kernel_launch(void* const* d_in, const int* in_sizes, int n_in,
                              void* d_out, int out_size, void* d_ws, size_t ws_size,
                              hipStream_t stream) {
  (void)in_sizes; (void)n_in; (void)d_ws; (void)ws_size; (void)out_size;
  const float* Zv = (const float*)d_in[0];  // [65536, 512, 2] fp32
  const float* Zp = (const float*)d_in[1];  // [65536, 512]    fp32
  float* out = (float*)d_out;               // S[65536,513] then V[65536,513]
  heston_paths_kernel<<<kBatch / kBlock, kBlock, 0, stream>>>(Zv, Zp, out);
}